// ContactMapPredictor_1872605741854
// MI455X (gfx1250) — compile-verified
//
#include <hip/hip_runtime.h>
#include <stdint.h>

typedef __attribute__((ext_vector_type(2))) float v2f;
typedef __attribute__((ext_vector_type(8))) float v8f;

#define D_DIM  128
#define B_DIM  2
#define N1_DIM 500
#define N1_PAD 512
#define N2_DIM 800

// ---------------------------------------------------------------------------
// Kernel 1: Q[b, row, col] = sum_k H[b,row,k] * W[k,col] (+ bias[col])
// One wave per 16x16 output tile, V_WMMA_F32_16X16X4_F32, K stepped by 4.
// fp32 end-to-end (matches the f32 reference exactly).
// ---------------------------------------------------------------------------
__global__ __launch_bounds__(32)
void proj_wmma_kernel(const float* __restrict__ H,   // [B, Nrows, D]
                      const float* __restrict__ W,   // [D, D] row-major
                      const float* __restrict__ bias,// [D] or nullptr
                      float* __restrict__ Q,         // [B, Npad, D]
                      int Nrows, int Npad)
{
    const int lane = threadIdx.x;
    const int half = lane >> 4;        // 0 or 1
    const int lr   = lane & 15;
    const int colBase = blockIdx.x * 16;
    const int rowBase = blockIdx.y * 16;
    const int b       = blockIdx.z;

    const float* Hb = H + (size_t)b * Nrows * D_DIM;
    const int row  = rowBase + lr;
    const int rowc = (row < Nrows) ? row : (Nrows - 1);   // clamp: keeps EXEC full
    const float* arow = Hb + (size_t)rowc * D_DIM + 2 * half;
    const float* bcol = W + (size_t)(2 * half) * D_DIM + colBase + lr;

    v8f acc = {};
#pragma unroll 4
    for (int k = 0; k < D_DIM; k += 4) {
        v2f a, bm;
        a.x  = arow[k];
        a.y  = arow[k + 1];
        bm.x = bcol[(size_t)k * D_DIM];
        bm.y = bcol[(size_t)(k + 1) * D_DIM];
        acc = __builtin_amdgcn_wmma_f32_16x16x4_f32(
            /*neg_a=*/false, a, /*neg_b=*/false, bm,
            /*c_mod=*/(short)0, acc, /*reuse_a=*/false, /*reuse_b=*/false);
    }

    const float bv = bias ? bias[colBase + lr] : 0.0f;
    float* Qb = Q + (size_t)b * Npad * D_DIM;
#pragma unroll
    for (int i = 0; i < 8; ++i) {
        const int rout = rowBase + i + 8 * half;   // always < Npad by construction
        Qb[(size_t)rout * D_DIM + colBase + lr] = acc[i] + bv;
    }
}

// ---------------------------------------------------------------------------
// Kernel 2: fused pairwise pass. Never materializes hidden[B,N1,N2,D].
//   pred[b,n,m]  = sum_d relu(q1[b,n,d] + q2[b,m,d]) * W2[d] + b2
//   mask[b,n,m]  = (S1!=0)*(S2!=0);  y = (cm<0.5)*mask
// 256 threads = one 16x16 (n,m) tile. Tiles staged straight into LDS via
// GLOBAL_LOAD_ASYNC_TO_LDS_B128 (ASYNCcnt path, no VGPR round-trip); the
// per-pair contact-map / sequence-mask loads are issued while the wave waits
// on ASYNCcnt so their latency hides under staging. LDS rows padded to 132
// floats so the 16 ml-lanes' float4 reads stagger by 4 banks (conflict-free).
// ---------------------------------------------------------------------------
__global__ __launch_bounds__(256)
void pair_pred_kernel(const float* __restrict__ Q1,  // [B, 512, D]
                      const float* __restrict__ Q2,  // [B, 800, D]
                      const int*   __restrict__ S1,  // [B, N1]
                      const int*   __restrict__ S2,  // [B, N2]
                      const float* __restrict__ CM,  // [B, N1, N2]
                      const float* __restrict__ W2,  // [D]
                      const float* __restrict__ b2,  // [1]
                      float* __restrict__ out)       // pred | y | mask
{
    __shared__ __align__(16) float q1s[16][132];
    __shared__ __align__(16) float q2s[16][132];
    __shared__ __align__(16) float w2s[D_DIM];

    const int tid   = threadIdx.x;
    const int mt    = blockIdx.x;
    const int nt    = blockIdx.y;
    const int b     = blockIdx.z;
    const int nbase = nt * 16;
    const int mbase = mt * 16;

    if (tid < D_DIM) w2s[tid] = W2[tid];

    // --- async stage both tiles: 16 rows x 128 floats each; 8 floats/thread ---
    {
        const int r = tid >> 4;          // row 0..15
        const int c = (tid & 15) << 3;   // col 0,8,...,120 (two b128 per thread)
        const float* g1 = Q1 + ((size_t)b * N1_PAD + nbase + r) * D_DIM + c;
        const float* g2 = Q2 + ((size_t)b * N2_DIM + mbase + r) * D_DIM + c;
        const uint32_t l1 = (uint32_t)(uintptr_t)&q1s[r][c];
        const uint32_t l2 = (uint32_t)(uintptr_t)&q2s[r][c];
        // INST_OFFSET applies to BOTH the global and LDS address (ISA §08 §4.4),
        // so offset:16 fetches the second float4 of this thread's chunk.
        asm volatile("global_load_async_to_lds_b128 %0, %1, off\n\t"
                     "global_load_async_to_lds_b128 %0, %1, off offset:16"
                     :: "v"(l1), "v"((uint64_t)(uintptr_t)g1) : "memory");
        asm volatile("global_load_async_to_lds_b128 %0, %1, off\n\t"
                     "global_load_async_to_lds_b128 %0, %1, off offset:16"
                     :: "v"(l2), "v"((uint64_t)(uintptr_t)g2) : "memory");
    }

    const int nl = tid >> 4;
    const int ml = tid & 15;
    const int n  = nbase + nl;
    const int m  = mbase + ml;

    // Issue the per-pair global loads now; latency hides under the async wait.
    float cmv = 1.0f;
    int   s1v = 0, s2v = 0;
    if (n < N1_DIM) {
        cmv = CM[((size_t)b * N1_DIM + n) * N2_DIM + m];
        s1v = S1[b * N1_DIM + n];
        s2v = S2[b * N2_DIM + m];
    }
    const float b2v = b2[0];

    asm volatile("s_wait_asynccnt 0x0" ::: "memory");
    __syncthreads();

    float acc = 0.0f;
#pragma unroll
    for (int d = 0; d < D_DIM; d += 4) {
        const float4 a = *reinterpret_cast<const float4*>(&q1s[nl][d]);
        const float4 q = *reinterpret_cast<const float4*>(&q2s[ml][d]);
        const float4 w = *reinterpret_cast<const float4*>(&w2s[d]);
        acc = fmaf(fmaxf(a.x + q.x, 0.0f), w.x, acc);
        acc = fmaf(fmaxf(a.y + q.y, 0.0f), w.y, acc);
        acc = fmaf(fmaxf(a.z + q.z, 0.0f), w.z, acc);
        acc = fmaf(fmaxf(a.w + q.w, 0.0f), w.w, acc);
    }

    if (n < N1_DIM) {
        const float pred = acc + b2v;
        const float mk   = ((s1v != 0) ? 1.0f : 0.0f) * ((s2v != 0) ? 1.0f : 0.0f);
        const float y    = (cmv < 0.5f) ? mk : 0.0f;
        const size_t plane = (size_t)B_DIM * N1_DIM * N2_DIM;   // 800000
        const size_t base  = (size_t)b * (N1_DIM * N2_DIM)
                           + (size_t)n * N2_DIM + m;
        out[base]             = pred;
        out[plane + base]     = y;
        out[2 * plane + base] = mk;
    }
}

extern "C" void kernel_launch(void* const* d_in, const int* in_sizes, int n_in,
                              void* d_out, int out_size, void* d_ws, size_t ws_size,
                              hipStream_t stream)
{
    (void)in_sizes; (void)n_in; (void)out_size; (void)ws_size;

    const int*   S1 = (const int*)  d_in[0];   // [B,N1] int32
    const int*   S2 = (const int*)  d_in[1];   // [B,N2] int32
    const float* h1 = (const float*)d_in[2];   // [B,N1,D]
    const float* h2 = (const float*)d_in[3];   // [B,N2,D]
    const float* cm = (const float*)d_in[4];   // [B,N1,N2]
    const float* W1 = (const float*)d_in[5];   // [2D,D]
    const float* b1 = (const float*)d_in[6];   // [D]
    const float* W2 = (const float*)d_in[7];   // [D,1]
    const float* b2 = (const float*)d_in[8];   // [1]

    float* q1 = (float*)d_ws;                               // [2,512,128]
    float* q2 = q1 + (size_t)B_DIM * N1_PAD * D_DIM;        // [2,800,128]

    // q1 = h1 @ W1[:128] + b1   (N1 padded to 512; clamped rows fill pad)
    proj_wmma_kernel<<<dim3(8, N1_PAD / 16, B_DIM), 32, 0, stream>>>(
        h1, W1, b1, q1, N1_DIM, N1_PAD);
    // q2 = h2 @ W1[128:]        (no bias; 800 % 16 == 0)
    proj_wmma_kernel<<<dim3(8, N2_DIM / 16, B_DIM), 32, 0, stream>>>(
        h2, W1 + D_DIM * D_DIM, nullptr, q2, N2_DIM, N2_DIM);
    // fused pairwise pred / y_contact / mask
    pair_pred_kernel<<<dim3(N2_DIM / 16, N1_PAD / 16, B_DIM), 256, 0, stream>>>(
        q1, q2, S1, S2, cm, W2, b2, (float*)d_out);
}